// GraphRegression_78400333021807
// MI455X (gfx1250) — compile-verified
//
#include <hip/hip_runtime.h>
#include <hip/hip_bf16.h>

typedef __attribute__((ext_vector_type(2))) float v2f;
typedef __attribute__((ext_vector_type(8))) float v8f;

#define HDIM 128

// ---------------- zero workspace ----------------
__global__ void zero_kernel(float4* __restrict__ p, int n4) {
    int i = blockIdx.x * blockDim.x + threadIdx.x;
    if (i < n4) p[i] = make_float4(0.f, 0.f, 0.f, 0.f);
}

// ---------------- degree accumulation ----------------
__global__ void degree_kernel(const int* __restrict__ src, const int* __restrict__ dst,
                              float* __restrict__ deg_out, float* __restrict__ deg_in, int E) {
    int e = blockIdx.x * blockDim.x + threadIdx.x;
    if (e < E) {
        atomicAdd(deg_out + src[e], 1.0f);
        atomicAdd(deg_in  + dst[e], 1.0f);
    }
}

// deg -> rsqrt(max(deg,1)) in place, both arrays
__global__ void invsqrt_kernel(float* __restrict__ a, float* __restrict__ b, int n) {
    int i = blockIdx.x * blockDim.x + threadIdx.x;
    if (i < n) {
        a[i] = rsqrtf(fmaxf(a[i], 1.0f));
        b[i] = rsqrtf(fmaxf(b[i], 1.0f));
    }
}

// ---------------- edge scatter-add: agg[dst] += x[src] * iso[src] ----------------
template<int D>
__global__ void edge_agg_kernel(const float* __restrict__ x, const int* __restrict__ src,
                                const int* __restrict__ dst, const float* __restrict__ iso,
                                float* __restrict__ agg, int E) {
    constexpr int LPE = D / 4;               // lanes per edge (float4 chunks)
    int t = blockIdx.x * blockDim.x + threadIdx.x;
    int e = t / LPE;
    int part = t % LPE;
    if (e >= E) return;
    int s = src[e];
    int d = dst[e];
    float sc = iso[s];
    const float4 v = *(const float4*)(x + (size_t)s * D + part * 4);
    float* o = agg + (size_t)d * D + part * 4;
    atomicAdd(o + 0, v.x * sc);
    atomicAdd(o + 1, v.y * sc);
    atomicAdd(o + 2, v.z * sc);
    atomicAdd(o + 3, v.w * sc);
}

// ---------------- WMMA fp32 GEMM: out = relu(A[n,K] @ W[K,128] * isi[:,None] + bias) ----------------
// One wave per 16x16 output tile. Block = 256 threads = 8 waves = one 16-row strip
// of all 128 columns. K is 64 or 128; n is a multiple of 16 -> EXEC all-1s, no
// predication anywhere (WMMA requirement).
template<int K>
__global__ void gemm_wmma_kernel(const float* __restrict__ A, const float* __restrict__ W,
                                 const float* __restrict__ bias, const float* __restrict__ isi,
                                 float* __restrict__ out) {
    const int lane = threadIdx.x & 31;
    const int wave = threadIdx.x >> 5;       // 0..7 -> column tile
    const int half = lane >> 4;              // 0 or 1
    const int ml   = lane & 15;
    const int row0 = blockIdx.x * 16;
    const int col0 = wave * 16;

    v8f acc = {};
    const float* arow = A + (size_t)(row0 + ml) * K;   // A row for this lane's M
    #pragma unroll
    for (int k0 = 0; k0 < K; k0 += 4) {
        const int ka = k0 + 2 * half;
        v2f a, b;
        a.x = arow[ka];                                   // A[m][ka]
        a.y = arow[ka + 1];                               // A[m][ka+1]
        b.x = W[(size_t)ka * HDIM + col0 + ml];           // B[ka][col]
        b.y = W[(size_t)(ka + 1) * HDIM + col0 + ml];     // B[ka+1][col]
        // D = A(16x4) x B(4x16) + C  -> v_wmma_f32_16x16x4_f32
        acc = __builtin_amdgcn_wmma_f32_16x16x4_f32(
            /*neg_a=*/false, a, /*neg_b=*/false, b,
            /*c_mod=*/(short)0, acc, /*reuse_a=*/false, /*reuse_b=*/false);
    }

    const int col = col0 + ml;
    const float bcol = bias[col];
    #pragma unroll
    for (int r = 0; r < 8; ++r) {
        const int row = row0 + r + 8 * half;             // C/D layout: VGPR r, half-wave
        float v = acc[r] * isi[row] + bcol;
        v = fmaxf(v, 0.0f);                              // relu (layer1) / outer relu (layer2)
        out[(size_t)row * HDIM + col] = v;
    }
}

// ---------------- per-graph node count ----------------
__global__ void count_kernel(const int* __restrict__ gid, float* __restrict__ gcnt, int n) {
    int i = blockIdx.x * blockDim.x + threadIdx.x;
    if (i < n) atomicAdd(gcnt + gid[i], 1.0f);
}

// ---------------- per-graph feature sums (graph_ids sorted -> flush on change) ----------------
__global__ void readout_kernel(const float* __restrict__ hf, const int* __restrict__ gid,
                               float* __restrict__ gsum, int n) {
    const int f = threadIdx.x;               // 0..127, one feature column per thread
    int start = blockIdx.x * 128;
    int end = start + 128; if (end > n) end = n;
    float acc = 0.0f;
    int curg = -1;
    for (int i = start; i < end; ++i) {
        int g = gid[i];
        if (g != curg) {
            if (curg >= 0) atomicAdd(gsum + (size_t)curg * HDIM + f, acc);
            acc = 0.0f;
            curg = g;
        }
        acc += hf[(size_t)i * HDIM + f];
    }
    if (curg >= 0) atomicAdd(gsum + (size_t)curg * HDIM + f, acc);
}

// ---------------- final: out[g] = (gsum[g,:]/cnt[g]) @ Wr + br ----------------
__global__ void final_kernel(const float* __restrict__ gsum, const float* __restrict__ gcnt,
                             const float* __restrict__ Wr, const float* __restrict__ br,
                             float* __restrict__ out) {
    const int g = blockIdx.x;
    const int f = threadIdx.x;               // 128 threads
    const float c = fmaxf(gcnt[g], 1.0f);
    float v = (gsum[(size_t)g * HDIM + f] / c) * Wr[f];
    #pragma unroll
    for (int off = 16; off > 0; off >>= 1) v += __shfl_down(v, off, 32);
    __shared__ float red[4];
    if ((f & 31) == 0) red[f >> 5] = v;
    __syncthreads();
    if (f == 0) out[g] = red[0] + red[1] + red[2] + red[3] + br[0];
}

extern "C" void kernel_launch(void* const* d_in, const int* in_sizes, int n_in,
                              void* d_out, int out_size, void* d_ws, size_t ws_size,
                              hipStream_t stream) {
    // inputs (setup_inputs order)
    const float* h   = (const float*)d_in[0];   // [N,64]
    const int*   src = (const int*)  d_in[1];   // [E]
    const int*   dst = (const int*)  d_in[2];   // [E]
    const int*   gid = (const int*)  d_in[3];   // [N] (sorted)
    const float* W1  = (const float*)d_in[5];   // [64,128]
    const float* b1  = (const float*)d_in[6];   // [128]
    const float* W2  = (const float*)d_in[7];   // [128,128]
    const float* b2  = (const float*)d_in[8];   // [128]
    const float* Wr  = (const float*)d_in[9];   // [128,1]
    const float* br  = (const float*)d_in[10];  // [1]
    float* out = (float*)d_out;

    const int N  = in_sizes[3];
    const int E  = in_sizes[1];
    const int IN = in_sizes[0] / N;             // 64
    const int G  = out_size;                    // 64

    // workspace layout (floats); hf aliases agg1 (agg1/h1 dead by GEMM2)
    float* ws      = (float*)d_ws;
    float* deg_out = ws;                                  // N   (-> iso)
    float* deg_in  = deg_out + N;                         // N   (-> isi)
    float* agg1    = deg_in  + N;                         // N*IN
    float* h1      = agg1 + (size_t)N * IN;               // N*H
    float* agg2    = h1   + (size_t)N * HDIM;             // N*H
    float* gsum    = agg2 + (size_t)N * HDIM;             // G*H
    float* gcnt    = gsum + (size_t)G * HDIM;             // G
    float* hf      = agg1;                                // N*H (reuse)
    const size_t tot_floats = (size_t)2 * N + (size_t)N * IN + (size_t)2 * N * HDIM
                              + (size_t)G * HDIM + G;

    // 1. zero all accumulators (whole span; multiple of 4 floats)
    {
        int n4 = (int)(tot_floats / 4);
        zero_kernel<<<(n4 + 255) / 256, 256, 0, stream>>>((float4*)ws, n4);
    }
    // 2. degrees
    degree_kernel<<<(E + 255) / 256, 256, 0, stream>>>(src, dst, deg_out, deg_in, E);
    // 3. iso = rsqrt(max(deg_out,1)), isi = rsqrt(max(deg_in,1)) in place
    invsqrt_kernel<<<(N + 255) / 256, 256, 0, stream>>>(deg_out, deg_in, N);
    // 4. layer1 edge aggregation: agg1[dst] += h[src] * iso[src]   (16 lanes/edge)
    {
        long long T = (long long)E * (IN / 4);
        edge_agg_kernel<64><<<(int)((T + 255) / 256), 256, 0, stream>>>(h, src, dst, deg_out, agg1, E);
    }
    // 5. h1 = relu(agg1 @ W1 * isi + b1)   — WMMA f32 16x16x4
    gemm_wmma_kernel<64><<<N / 16, 256, 0, stream>>>(agg1, W1, b1, deg_in, h1);
    // 6. layer2 edge aggregation: agg2[dst] += h1[src] * iso[src]  (32 lanes/edge)
    {
        long long T = (long long)E * (HDIM / 4);
        edge_agg_kernel<128><<<(int)((T + 255) / 256), 256, 0, stream>>>(h1, src, dst, deg_out, agg2, E);
    }
    // 7. hf = relu(agg2 @ W2 * isi + b2)   (fuses layer-2 epilogue with outer relu)
    gemm_wmma_kernel<128><<<N / 16, 256, 0, stream>>>(agg2, W2, b2, deg_in, hf);
    // 8. per-graph counts + sums
    count_kernel<<<(N + 255) / 256, 256, 0, stream>>>(gid, gcnt, N);
    readout_kernel<<<(N + 127) / 128, 128, 0, stream>>>(hf, gid, gsum, N);
    // 9. final projection
    final_kernel<<<G, 128, 0, stream>>>(gsum, gcnt, Wr, br, out);
}